// GraphPotts_40819369181402
// MI455X (gfx1250) — compile-verified
//
#include <hip/hip_runtime.h>
#include <hip/hip_bf16.h>

#define PB 2
#define PN 4096
#define PK 48
#define PQ 20

typedef float v2f __attribute__((ext_vector_type(2)));
typedef float v8f __attribute__((ext_vector_type(8)));

static __device__ __forceinline__ v8f wmma_f32_16x16x4(v2f a, v2f b, v8f c) {
  return __builtin_amdgcn_wmma_f32_16x16x4_f32(false, a, false, b,
                                               (short)0, c, false, false);
}

// One wave32 per node (b,i).
//   q = 0..15  : J_i[q] via V_WMMA_F32_16X16X4_F32 with one-hot B (exact f32,
//                A tiles consume J rows 0..15 exactly once, fully dense).
//   q = 16..19 : direct per-lane gather of the selected column element
//                (same cache lines as a dense read of rows 16..19; branchless).
// Software-pipelined: loads for k+1 are issued before the WMMAs for k.
__global__ __launch_bounds__(32) void potts_main_kernel(
    const int* __restrict__ S, const float* __restrict__ h,
    const float* __restrict__ J, const int* __restrict__ edge_idx,
    float* __restrict__ Ui, float* __restrict__ part) {
  const int node = blockIdx.x;        // node = b*PN + i
  const int lane = threadIdx.x;       // 0..31
  const int b = node >> 12;           // node / PN

  __shared__ int   sj[PK];
  __shared__ float jis[PQ];

  // Gather neighbor states S_j[k] = S[b, edge_idx[b,i,k]]
  {
    int e = edge_idx[(size_t)node * PK + lane];
    sj[lane] = S[b * PN + e];
  }
  if (lane < PK - 32) {
    int k = 32 + lane;
    int e = edge_idx[(size_t)node * PK + k];
    sj[k] = S[b * PN + e];
  }
  __syncthreads();  // single-wave workgroup: nop barrier + DScnt wait

  const float* Jnode = J + (size_t)node * (PK * PQ * PQ);
  const int row  = lane & 15;              // A-matrix M row for this lane
  const int sub  = (lane >= 16) ? 2 : 0;   // K-pair 0/1 vs 2/3 (A and B layouts)
  const int rrow = 16 + (lane & 3);        // tail row handled by this lane group

  v8f   c0   = {};    // D rows q = 0..15 (all 16 N columns identical)
  float racc = 0.0f;  // tail accumulator: J_i[16 + (lane&3)] in lanes 0..3

  // ---- pipeline prologue: operands for k = 0 ----
  v2f a[5];
#pragma unroll
  for (int c = 0; c < 5; ++c)
    a[c] = *(const v2f*)(Jnode + row * PQ + c * 4 + sub);
  int s = sj[0];

  // ---- steady state: issue k+1 loads, then compute k ----
#pragma unroll 2
  for (int k = 0; k < PK - 1; ++k) {
    const float* Jk  = Jnode + k * (PQ * PQ);
    const float* Jk1 = Jk + PQ * PQ;
    __builtin_prefetch(Jk1 + PQ * PQ, 0, 1);  // two blocks ahead -> GL2

    v2f an[5];
#pragma unroll
    for (int c = 0; c < 5; ++c)
      an[c] = *(const v2f*)(Jk1 + row * PQ + c * 4 + sub);
    int snext = sj[k + 1];

    // tail gather for current k (4 distinct addresses across the wave)
    float rj = Jk[rrow * PQ + s];

#pragma unroll
    for (int c = 0; c < 5; ++c) {
      const int k0 = c * 4 + sub;
      v2f bb;
      bb.x = (k0 == s)     ? 1.0f : 0.0f;
      bb.y = (k0 + 1 == s) ? 1.0f : 0.0f;
      c0 = wmma_f32_16x16x4(a[c], bb, c0);
    }
    racc += rj;

#pragma unroll
    for (int c = 0; c < 5; ++c) a[c] = an[c];
    s = snext;
  }

  // ---- epilogue: k = PK-1 ----
  {
    const float* Jk = Jnode + (PK - 1) * (PQ * PQ);
    float rj = Jk[rrow * PQ + s];
#pragma unroll
    for (int c = 0; c < 5; ++c) {
      const int k0 = c * 4 + sub;
      v2f bb;
      bb.x = (k0 == s)     ? 1.0f : 0.0f;
      bb.y = (k0 + 1 == s) ? 1.0f : 0.0f;
      c0 = wmma_f32_16x16x4(a[c], bb, c0);
    }
    racc += rj;
  }

  // C/D layout: VGPR r holds M=r (lanes 0-15) and M=r+8 (lanes 16-31), N=lane%16.
  // All N columns identical; read column N=0 from lanes 0 and 16.
  if (lane == 0) {
#pragma unroll
    for (int r = 0; r < 8; ++r) jis[r] = c0[r];
  }
  if (lane == 16) {
#pragma unroll
    for (int r = 0; r < 8; ++r) jis[8 + r] = c0[r];
  }
  if (lane < 4) jis[16 + lane] = racc;  // lanes 0..3 hold rows 16..19
  __syncthreads();

  if (lane < PQ) {
    float ji = jis[lane];
    float hv = h[(size_t)node * PQ + lane];
    Ui[(size_t)node * PQ + lane] = hv + ji;
    // U contribution: (h[s]+J_i[s]) - 0.5*J_i[s] = h[s] + 0.5*J_i[s]
    if (lane == S[node]) part[node] = hv + 0.5f * ji;
  }
}

// Deterministic fixed-order tree reduction of per-node contributions -> U[b].
__global__ __launch_bounds__(256) void potts_reduce_kernel(
    const float* __restrict__ part, float* __restrict__ U) {
  __shared__ float red[256];
  const int b = blockIdx.x;
  const int t = threadIdx.x;
  float s = 0.0f;
  for (int i = t; i < PN; i += 256) s += part[(size_t)b * PN + i];
  red[t] = s;
  __syncthreads();
#pragma unroll
  for (int off = 128; off > 0; off >>= 1) {
    if (t < off) red[t] += red[t + off];
    __syncthreads();
  }
  if (t == 0) U[b] = red[0];
}

extern "C" void kernel_launch(void* const* d_in, const int* in_sizes, int n_in,
                              void* d_out, int out_size, void* d_ws, size_t ws_size,
                              hipStream_t stream) {
  const int*   S        = (const int*)d_in[0];
  const float* h        = (const float*)d_in[1];
  const float* J        = (const float*)d_in[2];
  const int*   edge_idx = (const int*)d_in[3];

  // d_out: U (B floats) followed by U_i (B*N*Q floats), in return order.
  float* U    = (float*)d_out;
  float* Ui   = (float*)d_out + PB;
  float* part = (float*)d_ws;  // B*N floats = 32 KB scratch

  potts_main_kernel<<<dim3(PB * PN), dim3(32), 0, stream>>>(S, h, J, edge_idx, Ui, part);
  potts_reduce_kernel<<<dim3(PB), dim3(256), 0, stream>>>(part, U);
}